// GraphSage_33492154974614
// MI455X (gfx1250) — compile-verified
//
#include <hip/hip_runtime.h>

// ---------------- WMMA types (CDNA5 / gfx1250, wave32) ----------------
typedef __attribute__((ext_vector_type(2))) float v2f;
typedef __attribute__((ext_vector_type(8))) float v8f;
typedef __attribute__((ext_vector_type(4))) int   v4i;

#define D 128
#define TILE_ROWS 64
#define XS 260                       // padded X row stride (floats): rows 4 banks apart
#define HS 132                       // padded H row stride (floats): half-lanes 32 banks apart
#define X_FLOATS (TILE_ROWS * XS)    // 16640 floats = 65 KB
#define W_FLOATS (2 * D * D)         // 32768 floats = 128 KB (fragment-ordered)
#define SMEM_BYTES ((X_FLOATS + W_FLOATS) * 4)  // 197632 B < 320 KB/WGP

// ---------------- CDNA5 async global->LDS copies (ASYNCcnt path) -------
#define AS1 __attribute__((address_space(1)))
#define AS3 __attribute__((address_space(3)))

#if __has_builtin(__builtin_amdgcn_global_load_async_to_lds_b32)
#define HAVE_ASYNC_B32 1
#else
#define HAVE_ASYNC_B32 0
#endif
#if __has_builtin(__builtin_amdgcn_global_load_async_to_lds_b128)
#define HAVE_ASYNC_B128 1
#else
#define HAVE_ASYNC_B128 0
#endif

__device__ __forceinline__ void async_copy_b32(const float* g, float* l) {
#if HAVE_ASYNC_B32
    __builtin_amdgcn_global_load_async_to_lds_b32(
        (AS1 int*)(AS1 void*)g, (AS3 int*)(AS3 void*)l, 0, 0);
#else
    *l = *g;
#endif
}

__device__ __forceinline__ void async_copy_b128(const float* g, float* l) {
#if HAVE_ASYNC_B128
    __builtin_amdgcn_global_load_async_to_lds_b128(
        (AS1 v4i*)(AS1 void*)g, (AS3 v4i*)(AS3 void*)l, 0, 0);
#else
    *(float4*)l = *(const float4*)g;
#endif
}

__device__ __forceinline__ void wait_asynccnt0() {
#if __has_builtin(__builtin_amdgcn_s_wait_asynccnt)
    __builtin_amdgcn_s_wait_asynccnt(0);
#else
    asm volatile("s_wait_asynccnt 0" ::: "memory");
#endif
}

__device__ __forceinline__ float sigmoidf_fast(float x) {
    return 1.0f / (1.0f + __expf(-x));
}

// One SAGE layer: out[i] = l2norm(sigmoid(concat(Z[i], mean(Z[neigh[i]])) @ W + b))
// Block: 256 threads = 8 wave32; tile = 64 rows x 128 output cols.
// Wave w: row-tile (w&3), col-half (w>>2) -> four 16x16 WMMA accumulators.
template <int S>
__global__ __launch_bounds__(256, 1)
void sage_layer_kernel(const float* __restrict__ Zin,
                       const float* __restrict__ W,
                       const float* __restrict__ bias,
                       const int* __restrict__ neigh,
                       float* __restrict__ out,
                       int n)
{
    extern __shared__ float smem[];
    float* Xs = smem;                 // [64][XS]  concat features
    float* Ws = smem + X_FLOATS;      // W in B-fragment lane order
    float* Hs = smem;                 // [64][HS]  overlays Xs after GEMM

    const int tid  = threadIdx.x;
    const int lane = tid & 31;        // wave32
    const int wave = tid >> 5;        // 0..7
    const int half = lane >> 4;       // 0/1 (lane group)
    const int m16  = lane & 15;
    const int rowBase = blockIdx.x * TILE_ROWS;
    const float invs = 1.0f / (float)S;

    // ---- stage W into LDS, pre-swizzled into WMMA B-fragment order ----
    // B frag for (k-step k, col-tile ct): element (K=4k+rr, N=ct*16+nn) lives at
    // lane = (rr>=2?16:0)+nn, vgpr = rr&1  ->  Ws[((k*8+ct)*32+lane)*2+vgpr]
    // Issued as ASYNCcnt-tracked global->LDS copies so they overlap the gather.
    for (int i = tid; i < 2 * D * D; i += 256) {
        int r  = i >> 7;              // K row 0..255
        int c  = i & (D - 1);         // N col 0..127
        int k  = r >> 2;
        int rr = r & 3;
        int ct = c >> 4;
        int nn = c & 15;
        int dl = ((rr & 2) ? 16 : 0) + nn;
        int vg = rr & 1;
        int di = ((((k << 3) + ct) << 5) + dl) * 2 + vg;
        async_copy_b32(&W[i], &Ws[di]);
    }

    // ---- aggregate: X[r][0:128]=Z[row] (async copy), X[r][128:256]=mean(Z[neigh]) ----
    // wave w owns local rows w*8..w*8+7; each lane covers 4 columns (float4)
    for (int rr = 0; rr < 8; ++rr) {
        const int r  = wave * 8 + rr;
        const int gr = rowBase + r;
        float4 acc = make_float4(0.f, 0.f, 0.f, 0.f);
        if (gr < n) {
            // self row: pure copy global->LDS, one b128 per lane covers the row
            async_copy_b128(&Zin[(size_t)gr * D + lane * 4], &Xs[r * XS + lane * 4]);
            const int* nb = &neigh[(size_t)gr * S];
#pragma unroll
            for (int j = 0; j < S; ++j) {
                const int idx = nb[j];
                float4 v = *(const float4*)&Zin[(size_t)idx * D + lane * 4];
                acc.x += v.x; acc.y += v.y; acc.z += v.z; acc.w += v.w;
            }
        } else {
            *(float4*)&Xs[r * XS + lane * 4] = make_float4(0.f, 0.f, 0.f, 0.f);
        }
        float4 mn = make_float4(acc.x * invs, acc.y * invs, acc.z * invs, acc.w * invs);
        *(float4*)&Xs[r * XS + D + lane * 4] = mn;
    }

    wait_asynccnt0();   // W tiles + self rows resident in LDS for this wave
    __syncthreads();    // ... and for all waves

    // ---- GEMM via V_WMMA_F32_16X16X4_F32: K=256 in 64 steps ----
    const int rt = wave & 3;          // row tile
    const int ch = wave >> 2;         // col half (4 col tiles each)
    v8f acc0 = {}, acc1 = {}, acc2 = {}, acc3 = {};

    // A frag (16x4 f32): lanes 0-15 hold K=4k+{0,1}, lanes 16-31 K=4k+{2,3}
    const int arow = (rt * 16 + m16) * XS + 2 * half;
    const int lb   = lane << 1;
    const int cb   = ch << 8;         // ch*4 tiles * 64 floats/tile

    for (int k = 0; k < 64; ++k) {
        v2f a = *(const v2f*)&Xs[arow + (k << 2)];
        const float* wk = &Ws[(k << 9) + cb];
        v2f b0 = *(const v2f*)&wk[lb];
        v2f b1 = *(const v2f*)&wk[64 + lb];
        v2f b2 = *(const v2f*)&wk[128 + lb];
        v2f b3 = *(const v2f*)&wk[192 + lb];
        acc0 = __builtin_amdgcn_wmma_f32_16x16x4_f32(false, a, false, b0, (short)0, acc0, false, false);
        acc1 = __builtin_amdgcn_wmma_f32_16x16x4_f32(false, a, false, b1, (short)0, acc1, false, false);
        acc2 = __builtin_amdgcn_wmma_f32_16x16x4_f32(false, a, false, b2, (short)0, acc2, false, false);
        acc3 = __builtin_amdgcn_wmma_f32_16x16x4_f32(false, a, false, b3, (short)0, acc3, false, false);
    }

    __syncthreads();   // all waves done reading Xs before Hs overlays it

    // ---- epilogue 1: bias + sigmoid -> Hs ----
    // D-matrix layout: VGPR v holds rows {v, v+8}; col = tile*16 + (lane&15)
    {
        const int colBase = ch * 64;
        const int c0 = colBase + m16;
        const float bv0 = bias[c0];
        const float bv1 = bias[c0 + 16];
        const float bv2 = bias[c0 + 32];
        const float bv3 = bias[c0 + 48];
#pragma unroll
        for (int v = 0; v < 8; ++v) {
            const int row = rt * 16 + v + half * 8;
            float* hrow = &Hs[row * HS + c0];
            hrow[0]  = sigmoidf_fast(acc0[v] + bv0);
            hrow[16] = sigmoidf_fast(acc1[v] + bv1);
            hrow[32] = sigmoidf_fast(acc2[v] + bv2);
            hrow[48] = sigmoidf_fast(acc3[v] + bv3);
        }
    }

    __syncthreads();

    // ---- epilogue 2: row L2 norm + store (wave32 shuffle reduction) ----
    for (int rr = 0; rr < 8; ++rr) {
        const int r  = wave * 8 + rr;
        const int gr = rowBase + r;
        float4 h = *(const float4*)&Hs[r * HS + lane * 4];
        float ss = h.x * h.x + h.y * h.y + h.z * h.z + h.w * h.w;
        ss += __shfl_xor(ss, 16, 32);
        ss += __shfl_xor(ss, 8, 32);
        ss += __shfl_xor(ss, 4, 32);
        ss += __shfl_xor(ss, 2, 32);
        ss += __shfl_xor(ss, 1, 32);
        const float rn = rsqrtf(ss);
        if (gr < n) {
            float4 o = make_float4(h.x * rn, h.y * rn, h.z * rn, h.w * rn);
            *(float4*)&out[(size_t)gr * D + lane * 4] = o;
        }
    }
}

extern "C" void kernel_launch(void* const* d_in, const int* in_sizes, int n_in,
                              void* d_out, int out_size, void* d_ws, size_t ws_size,
                              hipStream_t stream) {
    const float* Z   = (const float*)d_in[0];
    const float* W0  = (const float*)d_in[1];
    const float* b0  = (const float*)d_in[2];
    const float* W1  = (const float*)d_in[3];
    const float* b1  = (const float*)d_in[4];
    const int*   nb0 = (const int*)d_in[5];
    const int*   nb1 = (const int*)d_in[6];

    const int N = in_sizes[0] / D;
    float* H0  = (float*)d_ws;          // layer-0 output: N*128 f32
    float* out = (float*)d_out;

    (void)n_in; (void)out_size; (void)ws_size;

    (void)hipFuncSetAttribute((const void*)sage_layer_kernel<25>,
                              hipFuncAttributeMaxDynamicSharedMemorySize, SMEM_BYTES);
    (void)hipFuncSetAttribute((const void*)sage_layer_kernel<10>,
                              hipFuncAttributeMaxDynamicSharedMemorySize, SMEM_BYTES);

    dim3 grid((N + TILE_ROWS - 1) / TILE_ROWS);
    dim3 block(256);

    sage_layer_kernel<25><<<grid, block, SMEM_BYTES, stream>>>(Z,  W0, b0, nb0, H0,  N);
    sage_layer_kernel<10><<<grid, block, SMEM_BYTES, stream>>>(H0, W1, b1, nb1, out, N);
}